// S4Layer_76888504533253
// MI455X (gfx1250) — compile-verified
//
#include <hip/hip_runtime.h>

// ---------------- problem constants ----------------
#define D_MODEL 512
#define D_STATE 64
#define BATCH   8
#define SEQ     2048
#define CHUNK   64
#define NCHUNK  (SEQ / CHUNK)       // 32
#define BT      (BATCH * SEQ)       // 16384

// ---------------- workspace layout (floats) ----------------
#define OFF_AD  0                                        // 64x64
#define OFF_BD  (OFF_AD + D_STATE*D_STATE)               // 64x512
#define OFF_P   (OFF_BD + D_STATE*D_MODEL)               // P[k-1]=A_d^k, k=1..64
#define OFF_V   (OFF_P  + CHUNK*D_STATE*D_STATE)         // v[bt][n]
#define OFF_X   (OFF_V  + (size_t)BT*D_STATE)            // x[bt][n]
#define OFF_XB  (OFF_X  + (size_t)BT*D_STATE)            // xb[b][c][n]
#define WS_FLOATS (OFF_XB + BATCH*NCHUNK*D_STATE)

typedef __attribute__((ext_vector_type(2))) float v2f;
typedef __attribute__((ext_vector_type(8))) float v8f;

// Native fp32 WMMA: D(16x16) = A(16x4) * B(4x16) + C  (confirmed lowering)
__device__ __forceinline__ v8f wmma4(v2f a, v2f b, v8f c) {
#if __has_builtin(__builtin_amdgcn_wmma_f32_16x16x4_f32)
  return __builtin_amdgcn_wmma_f32_16x16x4_f32(false, a, false, b, (short)0, c,
                                               false, false);
#else
  c[0] += a.x * b.x + a.y * b.y;
  return c;
#endif
}

// ---------------- CDNA5 async global->LDS staging (ASYNCcnt path) -------------
// 16 bytes per active lane. Builtin param0 is addrspace(1) int4* (per clang
// diagnostic: "int __vector_size__(4*sizeof(int)) __device__ *"); LDS side is
// the 32-bit LDS offset as an addrspace(3) pointer (aperture: addr[31:0]).
typedef int v4i __attribute__((vector_size(16)));
typedef __attribute__((address_space(1))) v4i as1_v4i;
typedef __attribute__((address_space(3))) v4i as3_v4i;

__device__ __forceinline__ void async_copy16(void* lds, const void* g) {
#if __has_builtin(__builtin_amdgcn_global_load_async_to_lds_b128)
  __builtin_amdgcn_global_load_async_to_lds_b128(
      (as1_v4i*)(unsigned long long)g,
      (as3_v4i*)(unsigned)(unsigned long long)lds, 0, 0);
#else
  *(float4*)lds = *(const float4*)g;      // synchronous fallback
#endif
}

__device__ __forceinline__ void wait_async0() {
#if __has_builtin(__builtin_amdgcn_s_wait_asynccnt)
  __builtin_amdgcn_s_wait_asynccnt(0);
#else
  asm volatile("s_wait_asynccnt 0x0" ::: "memory");
#endif
}

// ---------------- K1: dt reduction + discretization ----------------
__global__ void __launch_bounds__(512)
k_setup(const float* __restrict__ A, const float* __restrict__ Bm,
        const float* __restrict__ log_dt, float* __restrict__ ws) {
  __shared__ float red[512];
  __shared__ float sdt;
  int tid = threadIdx.x;
  red[tid] = expf(log_dt[tid]);               // D_MODEL == 512 == blockDim
  __syncthreads();
  for (int s = 256; s > 0; s >>= 1) {
    if (tid < s) red[tid] += red[tid + s];
    __syncthreads();
  }
  if (tid == 0) sdt = red[0] * (1.0f / 512.0f);
  __syncthreads();
  float dt = sdt;
  float* Ad = ws + OFF_AD;
  float* Bd = ws + OFF_BD;
  for (int i = tid; i < D_STATE * D_STATE; i += 512) {
    int r = i >> 6, c = i & 63;
    Ad[i] = dt * A[i] + (r == c ? 1.0f : 0.0f);
  }
  for (int i = tid; i < D_STATE * D_MODEL; i += 512) Bd[i] = dt * Bm[i];
}

// ---------------- K2: powers P_k = A_d^k (k=1..CHUNK) via WMMA chain ---------
#define LP 68   // padded LDS row stride (bank-conflict-free frag reads)
__global__ void __launch_bounds__(512)
k_powers(float* __restrict__ ws) {
  __shared__ __align__(16) float sAd[64 * LP];
  __shared__ __align__(16) float sP[2][64 * LP];
  const float* Ad = ws + OFF_AD;
  float* P = ws + OFF_P;
  int tid = threadIdx.x;
  for (int i = tid; i < 4096; i += 512) {
    int r = i >> 6, c = i & 63;
    float vv = Ad[i];
    sAd[r * LP + c] = vv;
    sP[0][r * LP + c] = vv;
    P[i] = vv;                                 // P_1 = A_d
  }
  __syncthreads();
  int lane = tid & 31, wid = tid >> 5;
  int mt = (wid >> 2) * 16, nt = (wid & 3) * 16;   // 16 waves -> 4x4 tiles
  int m = lane & 15, kh = lane >> 4;
  int pp = 0;
  for (int k = 2; k <= CHUNK; ++k) {
    const float* Pprev = sP[pp];
    float* Pnext = sP[pp ^ 1];
    v8f acc = {};
    #pragma unroll
    for (int kk = 0; kk < 16; ++kk) {
      int k0 = kk * 4 + 2 * kh;
      v2f a, b;
      a.x = Pprev[(mt + m) * LP + k0 + 0];     // A frag: row m, K = r + 2*kh
      a.y = Pprev[(mt + m) * LP + k0 + 1];
      b.x = sAd[(k0 + 0) * LP + nt + m];       // B frag: row K, col = lane
      b.y = sAd[(k0 + 1) * LP + nt + m];
      acc = wmma4(a, b, acc);
    }
    float* Pg = P + (size_t)(k - 1) * 4096;
    #pragma unroll
    for (int i = 0; i < 8; ++i) {              // C/D: row = i + 8*kh, col = lane
      int row = mt + i + 8 * kh;
      int col = nt + m;
      Pnext[row * LP + col] = acc[i];
      Pg[row * 64 + col] = acc[i];
    }
    __syncthreads();
    pp ^= 1;
  }
}

// ---------------- K3: v = B_d * u^T   (M=64, N=16384, K=512) -----------------
// Double-buffered: async-load K-chunk dk+1 into LDS while WMMAs consume dk.
#define LBD 36
__global__ void __launch_bounds__(512)
k_bdu(const float* __restrict__ u, float* __restrict__ ws) {
  __shared__ __align__(16) float sBd[2][64 * LBD];
  __shared__ __align__(16) float sU [2][64 * LBD];
  const float* Bd = ws + OFF_BD;
  float* v = ws + OFF_V;
  int tid = threadIdx.x;
  int bt0 = blockIdx.x * 64;                   // 64 (b,t) columns per block
  int lane = tid & 31, wid = tid >> 5;
  int mt = (wid >> 2) * 16, nt = (wid & 3) * 16;
  int m = lane & 15, kh = lane >> 4;
  int row = tid >> 3, dq = (tid & 7) * 4;      // one 16B transfer per thread

  // prologue: stage chunk 0 into buffer 0
  {
    int dbase = dq;
    async_copy16(&sBd[0][row * LBD + dq], &Bd[row * 512 + dbase]);
    async_copy16(&sU [0][row * LBD + dq], &u[(size_t)(bt0 + row) * 512 + dbase]);
  }
  wait_async0();
  __syncthreads();

  v8f acc = {};
  for (int dk = 0; dk < 16; ++dk) {            // K = 512 in chunks of 32
    int cur = dk & 1;
    if (dk + 1 < 16) {                         // prefetch next chunk (async)
      int dbase = (dk + 1) * 32 + dq;
      async_copy16(&sBd[cur ^ 1][row * LBD + dq], &Bd[row * 512 + dbase]);
      async_copy16(&sU [cur ^ 1][row * LBD + dq],
                   &u[(size_t)(bt0 + row) * 512 + dbase]);
    }
    #pragma unroll
    for (int kk = 0; kk < 8; ++kk) {
      int k0 = kk * 4 + 2 * kh;
      v2f a, b;
      a.x = sBd[cur][(mt + m) * LBD + k0];  a.y = sBd[cur][(mt + m) * LBD + k0 + 1];
      b.x = sU [cur][(nt + m) * LBD + k0];  b.y = sU [cur][(nt + m) * LBD + k0 + 1];
      acc = wmma4(a, b, acc);
    }
    wait_async0();                             // next buffer resident
    __syncthreads();                           // all waves done with cur
  }
  #pragma unroll
  for (int i = 0; i < 8; ++i) {
    int n  = mt + i + 8 * kh;
    int bt = bt0 + nt + m;
    v[(size_t)bt * 64 + n] = acc[i];
  }
}

// ---------------- K4: local chunk scans (256 independent scans) --------------
__global__ void __launch_bounds__(64)
k_scan(float* __restrict__ ws) {
  __shared__ float AdT[64 * 64];
  __shared__ float xs[64];
  const float* Ad = ws + OFF_AD;
  const float* v = ws + OFF_V;
  float* x = ws + OFF_X;
  int n = threadIdx.x;
  int b = blockIdx.x / NCHUNK, c = blockIdx.x % NCHUNK;
  for (int k = 0; k < 64; ++k) AdT[k * 64 + n] = Ad[n * 64 + k];
  xs[n] = 0.0f;
  __syncthreads();
  size_t base = ((size_t)b * SEQ + (size_t)c * CHUNK) * 64;
  for (int j = 0; j < CHUNK; ++j) {
    float acc = v[base + j * 64 + n];
    #pragma unroll
    for (int k = 0; k < 64; ++k) acc += AdT[k * 64 + n] * xs[k];
    __syncthreads();
    xs[n] = acc;
    __syncthreads();
    x[base + j * 64 + n] = acc;                // local state (zero-init chunk)
  }
}

// ---------------- K5: serial boundary propagation with P_CHUNK ---------------
__global__ void __launch_bounds__(64)
k_boundary(float* __restrict__ ws) {
  __shared__ float PLT[64 * 64];
  __shared__ float xs[64];
  const float* PL = ws + OFF_P + (size_t)(CHUNK - 1) * 4096;   // A_d^64
  const float* x = ws + OFF_X;
  float* xb = ws + OFF_XB;
  int n = threadIdx.x;
  int b = blockIdx.x;
  for (int k = 0; k < 64; ++k) PLT[k * 64 + n] = PL[n * 64 + k];
  float xc = x[((size_t)b * SEQ + CHUNK - 1) * 64 + n];
  xb[(b * NCHUNK + 0) * 64 + n] = xc;
  xs[n] = xc;
  __syncthreads();
  for (int c = 1; c < NCHUNK; ++c) {
    float acc = x[((size_t)b * SEQ + (size_t)c * CHUNK + CHUNK - 1) * 64 + n];
    #pragma unroll
    for (int k = 0; k < 64; ++k) acc += PLT[k * 64 + n] * xs[k];
    __syncthreads();
    xs[n] = acc;
    __syncthreads();
    xb[(b * NCHUNK + c) * 64 + n] = acc;       // true end-state of chunk c
  }
}

// ---------------- K6: correction x_t += P_{j+1} * xb[c-1] --------------------
__global__ void __launch_bounds__(64)
k_correct(float* __restrict__ ws) {
  __shared__ float PT[64 * 64];
  __shared__ float xbs[8 * 64];
  int n = threadIdx.x;
  int c = 1 + (int)blockIdx.x / CHUNK;
  int j = (int)blockIdx.x % CHUNK;
  const float* Pj = ws + OFF_P + (size_t)j * 4096;             // A_d^(j+1)
  const float* xb = ws + OFF_XB;
  float* x = ws + OFF_X;
  for (int k = 0; k < 64; ++k) PT[k * 64 + n] = Pj[n * 64 + k];
  for (int b = 0; b < BATCH; ++b)
    xbs[b * 64 + n] = xb[(b * NCHUNK + c - 1) * 64 + n];
  __syncthreads();
  for (int b = 0; b < BATCH; ++b) {
    float acc = 0.0f;
    #pragma unroll
    for (int k = 0; k < 64; ++k) acc += PT[k * 64 + n] * xbs[b * 64 + k];
    x[((size_t)b * SEQ + (size_t)c * CHUNK + j) * 64 + n] += acc;
  }
}

// ---------------- K7: y = C * x + D .* u   (M=512, N=16384, K=64) ------------
#define LO 68
__global__ void __launch_bounds__(512)
k_out(const float* __restrict__ u, const float* __restrict__ C,
      const float* __restrict__ Dv, float* __restrict__ ws,
      float* __restrict__ y) {
  __shared__ __align__(16) float sC[64 * LO];
  __shared__ __align__(16) float sX[64 * LO];
  const float* x = ws + OFF_X;
  int tid = threadIdx.x;
  int d0  = (blockIdx.x & 7) * 64;
  int bt0 = ((int)blockIdx.x >> 3) * 64;
  #pragma unroll
  for (int p = 0; p < 2; ++p) {                // async-stage 64x64 of C and x
    int lin = tid * 8 + p * 4;
    int r = lin >> 6, cq = lin & 63;
    async_copy16(&sC[r * LO + cq], &C[(size_t)d0 * 64 + lin]);
    async_copy16(&sX[r * LO + cq], &x[(size_t)bt0 * 64 + lin]);
  }
  wait_async0();
  __syncthreads();
  int lane = tid & 31, wid = tid >> 5;
  int mt = (wid >> 2) * 16, nt = (wid & 3) * 16;
  int m = lane & 15, kh = lane >> 4;
  v8f acc = {};
  #pragma unroll
  for (int kk = 0; kk < 16; ++kk) {
    int k0 = kk * 4 + 2 * kh;
    v2f a, b;
    a.x = sC[(mt + m) * LO + k0];  a.y = sC[(mt + m) * LO + k0 + 1];
    b.x = sX[(nt + m) * LO + k0];  b.y = sX[(nt + m) * LO + k0 + 1];
    acc = wmma4(a, b, acc);
  }
  #pragma unroll
  for (int i = 0; i < 8; ++i) {
    int d  = d0 + mt + i + 8 * kh;
    int bt = bt0 + nt + m;
    size_t ui = (size_t)bt * 512 + d;
    y[ui] = acc[i] + Dv[d] * u[ui];
  }
}

// ---------------- launcher ----------------
extern "C" void kernel_launch(void* const* d_in, const int* in_sizes, int n_in,
                              void* d_out, int out_size, void* d_ws, size_t ws_size,
                              hipStream_t stream) {
  (void)in_sizes; (void)n_in; (void)out_size;
  const float* u      = (const float*)d_in[0];
  const float* A      = (const float*)d_in[1];
  const float* B      = (const float*)d_in[2];
  const float* C      = (const float*)d_in[3];
  const float* Dv     = (const float*)d_in[4];
  const float* log_dt = (const float*)d_in[5];
  float* y  = (float*)d_out;
  float* ws = (float*)d_ws;
  if (ws_size < (size_t)WS_FLOATS * sizeof(float)) return;   // ~9.7 MB needed

  k_setup   <<<1,                 512, 0, stream>>>(A, B, log_dt, ws);
  k_powers  <<<1,                 512, 0, stream>>>(ws);
  k_bdu     <<<BT / 64,           512, 0, stream>>>(u, ws);
  k_scan    <<<BATCH * NCHUNK,     64, 0, stream>>>(ws);
  k_boundary<<<BATCH,              64, 0, stream>>>(ws);
  k_correct <<<(NCHUNK - 1) * CHUNK, 64, 0, stream>>>(ws);
  k_out     <<<8 * (BT / 64),     512, 0, stream>>>(u, C, Dv, ws, y);
}